// Monotonic_Attention_1838246003321
// MI455X (gfx1250) — compile-verified
//
#include <hip/hip_runtime.h>
#include <math.h>

typedef __attribute__((ext_vector_type(2))) float v2f;
typedef __attribute__((ext_vector_type(8))) float v8f;

#define SEQ 4096
#define DIM 1024
#define NB  16
#define SCHUNKS 8                   // s-split for context partials
#define SPERCHUNK (SEQ / SCHUNKS)   // 512

// ---------------------------------------------------------------------------
// Kernel 1: score[b,s] = dot(enc[b,s,:], dec[b,:]) via V_WMMA_F32_16X16X4_F32
// Each wave handles 16 consecutive s-rows. A = 16x4 fp32 tile of enc,
// B = 4x16 with only column N=0 holding dec_state, C accumulates 16x16;
// column 0 of C ends up holding the 16 scores.
// ---------------------------------------------------------------------------
__global__ __launch_bounds__(256) void score_kernel(
    const float* __restrict__ enc, const float* __restrict__ dec,
    float* __restrict__ scores) {
  const int b    = blockIdx.y;
  const int tid  = threadIdx.x;
  const int wave = tid >> 5;
  const int lane = tid & 31;
  const int m        = lane & 15;   // row within 16-row tile
  const int halfSel  = lane >> 4;   // 0: lanes 0-15, 1: lanes 16-31
  const bool isCol0  = (m == 0);    // lanes 0 and 16 feed B column N=0

  __shared__ __align__(16) float ldsDec[DIM];
  __shared__ __align__(16) float ldsA[8][16 * 68];   // stride 68 -> conflict-free

  // cooperative load of this batch's dec_state row (1024 f32)
  ((float4*)ldsDec)[tid] = ((const float4*)(dec + (size_t)b * DIM))[tid];
  __syncthreads();

  const int s0 = blockIdx.x * 128 + wave * 16;
  const float* encBase = enc + ((size_t)b * SEQ + s0) * DIM;

  v8f c = {};   // fp32 16x16 accumulator (8 VGPRs)

  for (int k0 = 0; k0 < DIM; k0 += 64) {
    // Stage a 16x64 fp32 tile into this wave's private LDS slice.
    // Per iteration the wave loads 2 rows x 64 floats (b128 per lane).
#pragma unroll
    for (int i = 0; i < 8; ++i) {
      const int row = 2 * i + halfSel;
      const int col = m * 4;
      float4 v = *(const float4*)(encBase + (size_t)row * DIM + k0 + col);
      *(float4*)(&ldsA[wave][row * 68 + col]) = v;
    }
    // 16 WMMA steps of K=4 cover the 64-wide chunk. Fully branchless:
    // every lane broadcast-reads the dec pair, then cndmask's it to zero
    // unless it sits in B column 0 (lane 0 -> K={0,1}, lane 16 -> K={2,3}).
#pragma unroll
    for (int kk = 0; kk < 16; ++kk) {
      v2f a;
      a.x = ldsA[wave][m * 68 + 4 * kk + 2 * halfSel + 0];
      a.y = ldsA[wave][m * 68 + 4 * kk + 2 * halfSel + 1];

      const float2 dv = *(const float2*)(&ldsDec[k0 + 4 * kk + 2 * halfSel]);
      v2f bb;
      bb.x = isCol0 ? dv.x : 0.0f;   // v_cndmask, no branch
      bb.y = isCol0 ? dv.y : 0.0f;

      c = __builtin_amdgcn_wmma_f32_16x16x4_f32(
          /*neg_a=*/false, a, /*neg_b=*/false, bb,
          /*c_mod=*/(short)0, c, /*reuse_a=*/false, /*reuse_b=*/false);
    }
  }

  // C layout: VGPR r, lanes 0-15 -> C[r, lane]; lanes 16-31 -> C[8+r, lane-16].
  // Column N=0 lives in lane 0 (rows 0..7) and lane 16 (rows 8..15).
  if (lane == 0 || lane == 16) {
    float* out = scores + (size_t)b * SEQ + s0 + halfSel * 8;
    out[0] = c[0]; out[1] = c[1]; out[2] = c[2]; out[3] = c[3];
    out[4] = c[4]; out[5] = c[5]; out[6] = c[6]; out[7] = c[7];
  }
}

// ---------------------------------------------------------------------------
// Kernel 2: per-batch scans along S.
//   cum[b,s]   = inclusive cumsum of prev_att
//   p          = sigmoid(score) * cum
//   excl[b,s]  = exclusive cumprod of (1-p)
//   att        = p * excl        -> written into d_out attention slot
// One block of 1024 threads per batch; 4 contiguous elements per thread.
// ---------------------------------------------------------------------------
__global__ __launch_bounds__(1024) void scan_kernel(
    const float* __restrict__ scores, const float* __restrict__ prev_att,
    float* __restrict__ att_out) {
  const int b = blockIdx.x;
  const int t = threadIdx.x;
  __shared__ __align__(16) float sd[1024];

  const float4 pa = ((const float4*)(prev_att + (size_t)b * SEQ))[t];
  const float cs1 = pa.x;
  const float cs2 = cs1 + pa.y;
  const float cs3 = cs2 + pa.z;
  const float cs4 = cs3 + pa.w;

  // block-wide inclusive add-scan of per-thread totals
  sd[t] = cs4;
  __syncthreads();
  for (int off = 1; off < 1024; off <<= 1) {
    float v = (t >= off) ? sd[t - off] : 0.0f;
    __syncthreads();
    sd[t] += v;
    __syncthreads();
  }
  const float addBase = (t == 0) ? 0.0f : sd[t - 1];

  const float4 sc = ((const float4*)(scores + (size_t)b * SEQ))[t];
  const float c1 = addBase + cs1, c2 = addBase + cs2;
  const float c3 = addBase + cs3, c4 = addBase + cs4;
  const float p1 = c1 / (1.0f + expf(-sc.x));
  const float p2 = c2 / (1.0f + expf(-sc.y));
  const float p3 = c3 / (1.0f + expf(-sc.z));
  const float p4 = c4 / (1.0f + expf(-sc.w));

  const float q1 = 1.0f - p1, q2 = 1.0f - p2, q3 = 1.0f - p3, q4 = 1.0f - p4;
  const float m1 = q1;
  const float m2 = m1 * q2;
  const float m3 = m2 * q3;
  const float m4 = m3 * q4;

  __syncthreads();   // all addBase reads of sd done before overwrite
  sd[t] = m4;
  __syncthreads();
  for (int off = 1; off < 1024; off <<= 1) {
    float v = (t >= off) ? sd[t - off] : 1.0f;
    __syncthreads();
    sd[t] *= v;
    __syncthreads();
  }
  const float mulBase = (t == 0) ? 1.0f : sd[t - 1];

  float4 att;
  att.x = p1 * mulBase;
  att.y = p2 * (mulBase * m1);
  att.z = p3 * (mulBase * m2);
  att.w = p4 * (mulBase * m3);
  ((float4*)(att_out + (size_t)b * SEQ))[t] = att;
}

// ---------------------------------------------------------------------------
// Kernel 3: partial context sums. Block (sc, b) accumulates 512 s-rows over
// all 1024 d (float4 per lane). Attention chunk staged in LDS.
// ---------------------------------------------------------------------------
__global__ __launch_bounds__(256) void context_partial_kernel(
    const float* __restrict__ enc, const float* __restrict__ att,
    float* __restrict__ partial) {
  const int b    = blockIdx.y;
  const int sc   = blockIdx.x;
  const int tid  = threadIdx.x;
  const int sOff = sc * SPERCHUNK;

  __shared__ __align__(16) float satt[SPERCHUNK];
  ((float2*)satt)[tid] = ((const float2*)(att + (size_t)b * SEQ + sOff))[tid];
  __syncthreads();

  const float* e = enc + ((size_t)b * SEQ + sOff) * DIM + tid * 4;
  float ax = 0.f, ay = 0.f, az = 0.f, aw = 0.f;
#pragma unroll 4
  for (int s = 0; s < SPERCHUNK; ++s) {
    const float a = satt[s];
    const float4 v = *(const float4*)(e + (size_t)s * DIM);
    ax += a * v.x; ay += a * v.y; az += a * v.z; aw += a * v.w;
  }
  float4 acc; acc.x = ax; acc.y = ay; acc.z = az; acc.w = aw;
  ((float4*)(partial + (size_t)(sc * NB + b) * DIM))[tid] = acc;
}

// ---------------------------------------------------------------------------
// Kernel 4: reduce the 8 partials per batch into the context slot of d_out.
// ---------------------------------------------------------------------------
__global__ __launch_bounds__(256) void context_reduce_kernel(
    const float* __restrict__ partial, float* __restrict__ ctx) {
  const int b   = blockIdx.x;
  const int tid = threadIdx.x;
  float ax = 0.f, ay = 0.f, az = 0.f, aw = 0.f;
#pragma unroll
  for (int sc = 0; sc < SCHUNKS; ++sc) {
    const float4 v = ((const float4*)(partial + (size_t)(sc * NB + b) * DIM))[tid];
    ax += v.x; ay += v.y; az += v.z; aw += v.w;
  }
  float4 acc; acc.x = ax; acc.y = ay; acc.z = az; acc.w = aw;
  ((float4*)(ctx + (size_t)b * DIM))[tid] = acc;
}

// ---------------------------------------------------------------------------
// Launcher. Outputs: d_out[0 .. B*D)       = context_vec
//                    d_out[B*D .. B*D+B*S) = attention
// ---------------------------------------------------------------------------
extern "C" void kernel_launch(void* const* d_in, const int* in_sizes, int n_in,
                              void* d_out, int out_size, void* d_ws, size_t ws_size,
                              hipStream_t stream) {
  const float* enc      = (const float*)d_in[0];   // [16,4096,1024]
  const float* dec      = (const float*)d_in[1];   // [16,1024]
  const float* prev_att = (const float*)d_in[2];   // [16,4096]

  float* ctx_out = (float*)d_out;                  // [16,1024]
  float* att_out = (float*)d_out + NB * DIM;       // [16,4096]

  float* scores  = (float*)d_ws;                   // 16*4096 f32 = 256 KB
  float* partial = scores + (size_t)NB * SEQ;      // 8*16*1024 f32 = 512 KB

  score_kernel<<<dim3(SEQ / 128, NB), 256, 0, stream>>>(enc, dec, scores);
  scan_kernel<<<NB, 1024, 0, stream>>>(scores, prev_att, att_out);
  context_partial_kernel<<<dim3(SCHUNKS, NB), 256, 0, stream>>>(enc, att_out, partial);
  context_reduce_kernel<<<NB, 256, 0, stream>>>(partial, ctx_out);
}